// SelfCorrelationProject_32719060861402
// MI455X (gfx1250) — compile-verified
//
#include <hip/hip_runtime.h>

// ---------------------------------------------------------------------------
// SelfCorrelationProject for MI455X (gfx1250, wave32).
//
//   feat = w_reduce(64x256) @ x      -> WMMA fp32 GEMM, B staged via
//                                       GLOBAL_LOAD_ASYNC_TO_LDS_B128
//   att  = softmax(cosine-sim over 7x7 window of feat)  -> VALU + LDS
//   corr = sum_k att_k * patch_k
//   out  = w_proj(512x64) @ corr     -> same WMMA GEMM (K=64)
//
// Roofline: ~100 MB mandatory HBM traffic (x in, out out) = ~4.3 us at
// 23.3 TB/s; only 1.6 GMAC of matrix work -> fp32 V_WMMA_F32_16X16X4_F32
// keeps exact fp32 precision while staying under the memory floor. The
// streaming B operand moves global->LDS asynchronously (ASYNCcnt) in 128-bit
// transactions; WMMA fragments are then fed from LDS (bank-conflict-free,
// row stride 80 words so the two half-wave rows land 32 banks apart).
// ---------------------------------------------------------------------------

typedef float v2f __attribute__((ext_vector_type(2)));
typedef float v8f __attribute__((ext_vector_type(8)));

#define WMMA_F32(a, b, c)                                                      \
  __builtin_amdgcn_wmma_f32_16x16x4_f32(false, (a), false, (b), (short)0, (c), \
                                        false, false)

// Y[M,N] = W[M,K] * X[K,N] (row-major; per-batch X/Y strides).
// Block = 128 threads (4 waves). Block tile: 64 rows x 64 cols of Y;
// each wave owns a 64x16 sub-tile (4 v8f accumulators).
// grid = (N/64, M/64, B).
template <int K, int N>
__global__ __launch_bounds__(128, 1) void gemm_wmma_f32_kernel(
    const float* __restrict__ Wm, const float* __restrict__ X,
    float* __restrict__ Y, long xbs, long ybs) {
  constexpr int KB = 16;        // K-rows staged into LDS per iteration
  constexpr int LDP = 80;       // padded LDS row stride (words)
  __shared__ float s_x[KB * LDP];  // 5 KB staging buffer for the B tile

  const int tid = threadIdx.x;
  const int lane = tid & 31;
  const int wave = tid >> 5;
  const int lo = lane & 15;  // M (A) / N (B) index within 16-wide tile
  const int hi = lane >> 4;  // selects K sub-pair {0,1} vs {2,3}
  const int nblk = blockIdx.x * 64;    // block's 64-column range
  const int ncol = wave * 16 + lo;     // this lane's column within block tile
  const int r0 = blockIdx.y * 64;
  const float* Xb = X + (long)blockIdx.z * xbs;
  float* Yb = Y + (long)blockIdx.z * ybs;

  // Async staging pattern: 128 threads x 16B = 8 rows x 64 cols per issue;
  // two issues cover the 16x64 tile.
  const int sr = tid >> 4;         // staging row 0..7 (and sr+8)
  const int sc = (tid & 15) * 4;   // staging column (float4 granularity)
  const unsigned ld0 = (unsigned)(sr * LDP + sc) * 4u;
  const unsigned ld1 = (unsigned)((sr + 8) * LDP + sc) * 4u;
  const unsigned lbase = (unsigned)(unsigned long long)(uintptr_t)&s_x[0];

  v8f acc0 = {}; v8f acc1 = {}; v8f acc2 = {}; v8f acc3 = {};

  for (int k0 = 0; k0 < K; k0 += KB) {
    // ---- stage B tile [k0..k0+15] x [nblk..nblk+63] into LDS (async) ----
    const float* src0 = Xb + (long)(k0 + sr) * N + nblk + sc;
    const float* src1 = src0 + (long)8 * N;
    asm volatile(
        "global_load_async_to_lds_b128 %0, %2, off\n\t"
        "global_load_async_to_lds_b128 %1, %3, off\n\t"
        "s_wait_asynccnt 0x0"
        :
        : "v"(lbase + ld0), "v"(lbase + ld1), "v"(src0), "v"(src1)
        : "memory");
    __syncthreads();

    // ---- 4 WMMA K-steps over the staged tile ----
#pragma unroll
    for (int kk = 0; kk < KB; kk += 4) {
      const int kw = kk + 2 * hi;
      v2f bv;
      bv.x = s_x[(kw + 0) * LDP + ncol];
      bv.y = s_x[(kw + 1) * LDP + ncol];
      const int ka = k0 + kw;
      v2f a0 = *(const v2f*)(Wm + (long)(r0 +  0 + lo) * K + ka);
      v2f a1 = *(const v2f*)(Wm + (long)(r0 + 16 + lo) * K + ka);
      v2f a2 = *(const v2f*)(Wm + (long)(r0 + 32 + lo) * K + ka);
      v2f a3 = *(const v2f*)(Wm + (long)(r0 + 48 + lo) * K + ka);
      acc0 = WMMA_F32(a0, bv, acc0);
      acc1 = WMMA_F32(a1, bv, acc1);
      acc2 = WMMA_F32(a2, bv, acc2);
      acc3 = WMMA_F32(a3, bv, acc3);
    }
    __syncthreads();  // LDS tile reuse fence
  }

  // C/D 16x16 layout: lanes 0-15 -> rows j, lanes 16-31 -> rows j+8
  const int col = nblk + ncol;
#pragma unroll
  for (int j = 0; j < 8; ++j) {
    const int row = j + 8 * hi;
    Yb[(long)(r0 +  0 + row) * N + col] = acc0[j];
    Yb[(long)(r0 + 16 + row) * N + col] = acc1[j];
    Yb[(long)(r0 + 32 + row) * N + col] = acc2[j];
    Yb[(long)(r0 + 48 + row) * N + col] = acc3[j];
  }
}

#define CR 64
#define WIN 7
#define KWIN 49
#define CORR_BLOCK 256

// One thread per pixel. Center feature vector (64 ch) lives in registers via
// fully unrolled channel loops; the 49 attention weights live in LDS so the
// dynamically-indexed softmax does not spill to scratch.
__global__ __launch_bounds__(CORR_BLOCK, 1) void corr_attention_kernel(
    const float* __restrict__ feat, float* __restrict__ corr, int B, int H,
    int Wd) {
  __shared__ float att_s[KWIN * CORR_BLOCK];

  const int HW = H * Wd;
  const int idx = blockIdx.x * blockDim.x + threadIdx.x;
  if (idx >= B * HW) return;
  const int tid = threadIdx.x;
  const int b = idx / HW;
  const int p = idx - b * HW;
  const int h = p / Wd;
  const int w = p - h * Wd;
  const float* fb = feat + (long)b * CR * HW;

  // center feature -> registers
  float fc[CR];
  float cn2 = 0.f;
#pragma unroll
  for (int c = 0; c < CR; ++c) {
    fc[c] = fb[(long)c * HW + p];
    cn2 += fc[c] * fc[c];
  }
  const float cn = sqrtf(cn2);

  // pass 1: cosine similarity per window offset (zero-padded OOB -> sim = 0)
  float mx = -1e30f;
#pragma unroll 1
  for (int k = 0; k < KWIN; ++k) {
    const int hh = h + (k / WIN) - (WIN / 2);
    const int ww = w + (k % WIN) - (WIN / 2);
    float sim = 0.f;
    if (hh >= 0 && hh < H && ww >= 0 && ww < Wd) {
      const long np = (long)hh * Wd + ww;
      float dot = 0.f, pn2 = 0.f;
#pragma unroll
      for (int c = 0; c < CR; ++c) {
        const float g = fb[(long)c * HW + np];
        dot += fc[c] * g;
        pn2 += g * g;
      }
      sim = dot / fmaxf(cn * sqrtf(pn2), 1e-8f);
    }
    att_s[k * CORR_BLOCK + tid] = sim;
    mx = fmaxf(mx, sim);
  }

  // softmax over the 49 window positions
  float s = 0.f;
#pragma unroll 1
  for (int k = 0; k < KWIN; ++k) {
    const float e = expf(att_s[k * CORR_BLOCK + tid] - mx);
    att_s[k * CORR_BLOCK + tid] = e;
    s += e;
  }
  const float inv = 1.f / s;

  // pass 2: weighted patch sum (64-channel accumulator in registers)
  float acc[CR];
#pragma unroll
  for (int c = 0; c < CR; ++c) acc[c] = 0.f;

#pragma unroll 1
  for (int k = 0; k < KWIN; ++k) {
    const int hh = h + (k / WIN) - (WIN / 2);
    const int ww = w + (k % WIN) - (WIN / 2);
    if (hh < 0 || hh >= H || ww < 0 || ww >= Wd) continue;  // zero patch
    const long np = (long)hh * Wd + ww;
    const float a = att_s[k * CORR_BLOCK + tid] * inv;
#pragma unroll
    for (int c = 0; c < CR; ++c) acc[c] += a * fb[(long)c * HW + np];
  }

  float* cb = corr + (long)b * CR * HW;
#pragma unroll
  for (int c = 0; c < CR; ++c) cb[(long)c * HW + p] = acc[c];
}

extern "C" void kernel_launch(void* const* d_in, const int* in_sizes, int n_in,
                              void* d_out, int out_size, void* d_ws,
                              size_t ws_size, hipStream_t stream) {
  const float* x = (const float*)d_in[0];         // [8,256,64,64]
  const float* w_reduce = (const float*)d_in[1];  // [64,256]
  const float* w_proj = (const float*)d_in[2];    // [512,64]
  float* out = (float*)d_out;                     // [8,512,64,64]

  const int B = 8, C = 256, H = 64, Wd = 64, CO = 512;
  const int HW = H * Wd;  // 4096

  float* feat = (float*)d_ws;                     // [8,64,64,64]  8.4 MB
  float* corr = feat + (size_t)B * CR * HW;       // [8,64,64,64]  8.4 MB

  // feat = w_reduce @ x : M=64, K=256, N=4096 per batch
  {
    dim3 grid(HW / 64, 64 / 64, B);
    gemm_wmma_f32_kernel<256, 4096><<<grid, 128, 0, stream>>>(
        w_reduce, x, feat, (long)C * HW, (long)CR * HW);
  }

  // windowed cosine-sim softmax attention
  {
    const int n = B * HW;
    corr_attention_kernel<<<(n + CORR_BLOCK - 1) / CORR_BLOCK, CORR_BLOCK, 0,
                            stream>>>(feat, corr, B, H, Wd);
  }

  // out = w_proj @ corr : M=512, K=64, N=4096 per batch
  {
    dim3 grid(HW / 64, CO / 64, B);
    gemm_wmma_f32_kernel<64, 4096><<<grid, 128, 0, stream>>>(
        w_proj, corr, out, (long)CR * HW, (long)CO * HW);
  }
}